// PoincareMLR_42064909697555
// MI455X (gfx1250) — compile-verified
//
#include <hip/hip_runtime.h>
#include <math.h>

#define FEAT_DIM 256
#define NUM_OUT  128
#define BATCH    2048
#define EPSV     1e-15f
#define LDA      264   // padded LDS row stride in dwords (264 mod 64 == 8)

typedef float v2f __attribute__((ext_vector_type(2)));
typedef float v8f __attribute__((ext_vector_type(8)));
typedef unsigned int v4u __attribute__((ext_vector_type(4)));
typedef int v8i __attribute__((ext_vector_type(8)));
typedef int v4i __attribute__((ext_vector_type(4)));

// Use the Tensor Data Mover for global->LDS staging when the builtin exists.
// This toolchain exposes the 6-arg form:
//   (uint32x4 g0, int32x8 g1, int32x4 g2, int32x4 g3, int32x8 g4, i32 cpol)
#if defined(__HIP_DEVICE_COMPILE__)
#  if __has_builtin(__builtin_amdgcn_tensor_load_to_lds)
#    define USE_TDM 1
#  endif
#endif
#ifndef USE_TDM
#  define USE_TDM 0
#endif

// ---------------------------------------------------------------------------
// Kernel 1: per-class precompute.
//   p_k = expmap0(point_k) = tanh(r)/r * point_k,  r = max(||point_k||, EPS)
//   P2[k] = ||p_k||^2, PT[k] = <p_k, tangent_k>, AN[k] = ||tangent_k||
// ---------------------------------------------------------------------------
__global__ void poincare_prep(const float* __restrict__ point,
                              const float* __restrict__ tangent,
                              float* __restrict__ P,
                              float* __restrict__ P2,
                              float* __restrict__ PT,
                              float* __restrict__ AN) {
  __shared__ float s0[FEAT_DIM], s1[FEAT_DIM], s2[FEAT_DIM];
  const int k = blockIdx.x;
  const int t = threadIdx.x;
  const float pv = point[k * FEAT_DIM + t];
  const float tv = tangent[k * FEAT_DIM + t];
  s0[t] = pv * pv;
  s1[t] = pv * tv;
  s2[t] = tv * tv;
  __syncthreads();
  for (int off = FEAT_DIM / 2; off > 0; off >>= 1) {
    if (t < off) {
      s0[t] += s0[t + off];
      s1[t] += s1[t + off];
      s2[t] += s2[t + off];
    }
    __syncthreads();
  }
  const float pn2 = s0[0], ptn = s1[0], tn2 = s2[0];
  const float r = fmaxf(sqrtf(pn2), EPSV);
  const float scale = tanhf(r) / r;
  P[k * FEAT_DIM + t] = scale * pv;
  if (t == 0) {
    P2[k] = scale * scale * pn2;
    PT[k] = scale * ptn;
    AN[k] = sqrtf(tn2);
  }
}

#if USE_TDM
// LDS byte offset of a generic pointer known to point into LDS.
__device__ inline unsigned lds_off(void* p) {
  return (unsigned)(unsigned long long)(__attribute__((address_space(3))) void*)p;
}

// Issue one TDM 2D tile load: rows x 256 fp32 tile, row-major source with
// rowStrideElems element stride, into LDS with a 264-dword padded row stride
// (pad_interval=7 -> pad after every 256 dwords; pad_amount=7 -> 8 dwords).
__device__ inline void tdm_load_2d(const float* gsrc, unsigned ldsByteOff,
                                   unsigned rows, unsigned rowStrideElems,
                                   unsigned tensorRows) {
  const unsigned long long ga = (unsigned long long)gsrc;

  v4u g0;
  g0.x = 1u;                                     // count=1 (valid user D#)
  g0.y = ldsByteOff;                             // lds_addr (bytes)
  g0.z = (unsigned)(ga & 0xFFFFFFFFu);           // global_addr[31:0]
  g0.w = (unsigned)((ga >> 32) & 0x01FFFFFFu)    // global_addr[56:32]
       | (2u << 30);                             // type = 2 ("image")

  v8i g1;
  g1[0] = (2 << 16)                              // data_size = 4 bytes
        | (1 << 20)                              // pad_enable
        | (7 << 22)                              // pad_interval: 256 dwords
        | (7 << 25);                             // pad_amount: 8 dwords
  g1[1] = (int)(256u << 16);                     // tensor_dim0[15:0] = 256
  g1[2] = (int)(tensorRows << 16);               // tensor_dim1[15:0]
  g1[3] = (int)(256u << 16);                     // tile_dim0 = 256
  g1[4] = (int)rows;                             // tile_dim1 = rows, tile_dim2=0
  g1[5] = (int)rowStrideElems;                   // tensor_dim0_stride[31:0]
  g1[6] = 0;                                     // stride hi / dim1_stride lo
  g1[7] = 0;

  const v4i gz4 = {0, 0, 0, 0};                  // 2D tensor: groups 2/3 unused
  const v8i gz8 = {0, 0, 0, 0, 0, 0, 0, 0};
  __builtin_amdgcn_tensor_load_to_lds(g0, g1, gz4, gz4, gz8, 0);
}
#endif

// ---------------------------------------------------------------------------
// Kernel 2: fused dual-GEMM (S = X P^T, T = X Tan^T) + hyperbolic epilogue.
// Block: 128 threads (4 wave32). Wave w owns a 16x16 tile at cols c0 + 16w.
// Staging: TDM (tensor_load_to_lds) with HW pad -> 264-dword LDS rows.
// Math:   v_wmma_f32_16x16x4_f32, K stepped by 4, fp32 end-to-end.
// ---------------------------------------------------------------------------
__global__ __launch_bounds__(128)
void poincare_mlr(const float* __restrict__ X,
                  const float* __restrict__ P,
                  const float* __restrict__ T,
                  const float* __restrict__ P2,
                  const float* __restrict__ PT,
                  const float* __restrict__ AN,
                  float* __restrict__ out) {
  extern __shared__ float smem[];
  float* ldsA = smem;                 // 16 x LDA  (X tile, 16 rows x 256)
  float* ldsP = smem + 16 * LDA;      // 64 x LDA  (scaled points panel)
  float* ldsT = ldsP + 64 * LDA;      // 64 x LDA  (tangent panel)

  const int tid  = threadIdx.x;       // 0..127
  const int wave = tid >> 5;          // 0..3
  const int lane = tid & 31;
  const int r0 = blockIdx.x * 16;     // batch-row tile
  const int c0 = blockIdx.y * 64;     // class-col strip

#if USE_TDM
  if (wave == 0) {                    // one wave issues the three DMAs
    tdm_load_2d(X + (size_t)r0 * FEAT_DIM, lds_off(ldsA), 16, FEAT_DIM, BATCH);
    tdm_load_2d(P + (size_t)c0 * FEAT_DIM, lds_off(ldsP), 64, FEAT_DIM, NUM_OUT);
    tdm_load_2d(T + (size_t)c0 * FEAT_DIM, lds_off(ldsT), 64, FEAT_DIM, NUM_OUT);
    __builtin_amdgcn_s_wait_tensorcnt(0);
  }
  __syncthreads();
#else
  for (int e = tid; e < 16 * FEAT_DIM; e += 128) {
    const int m = e >> 8, d = e & 255;
    ldsA[m * LDA + d] = X[(r0 + m) * FEAT_DIM + d];
  }
  for (int e = tid; e < 64 * FEAT_DIM; e += 128) {
    const int c = e >> 8, d = e & 255;
    ldsP[c * LDA + d] = P[(c0 + c) * FEAT_DIM + d];
    ldsT[c * LDA + d] = T[(c0 + c) * FEAT_DIM + d];
  }
  __syncthreads();
#endif

  // WMMA fragment indexing (16x4 A / 4x16 B striping over wave32):
  //   lanes 0-15 : row/col = lane,      hold K = {0,1}
  //   lanes 16-31: row/col = lane - 16, hold K = {2,3}
  const int hn = lane & 15;
  const int kh = (lane >> 4) * 2;
  const float* aBase = &ldsA[hn * LDA + kh];
  const float* pBase = &ldsP[(wave * 16 + hn) * LDA + kh];
  const float* tBase = &ldsT[(wave * 16 + hn) * LDA + kh];

  v8f accS = {};
  v8f accT = {};
#pragma unroll 4
  for (int d = 0; d < FEAT_DIM; d += 4) {
    const v2f a  = *(const v2f*)(aBase + d);
    const v2f bp = *(const v2f*)(pBase + d);
    const v2f bt = *(const v2f*)(tBase + d);
    accS = __builtin_amdgcn_wmma_f32_16x16x4_f32(false, a, false, bp,
                                                 (short)0, accS, false, false);
    accT = __builtin_amdgcn_wmma_f32_16x16x4_f32(false, a, false, bt,
                                                 (short)0, accT, false, false);
  }

  // Per-row ||x_b||^2: lane computes row hn's norm from the LDS A tile.
  float y2own = 0.f;
#pragma unroll 8
  for (int d = 0; d < FEAT_DIM; d += 2) {
    const v2f v = *(const v2f*)(&ldsA[hn * LDA + d]);
    y2own = fmaf(v.x, v.x, y2own);
    y2own = fmaf(v.y, v.y, y2own);
  }

  // Per-class scalars for this lane's output column.
  const int col = c0 + wave * 16 + hn;
  const float p2c = P2[col];
  const float ptc = PT[col];
  const float anc = AN[col];
  const float beta = 1.f - p2c;

  // Epilogue over the 8 accumulator rows: C/D layout is
  //   VGPR i, lanes 0-15 -> (m = i,   n = lane); lanes 16-31 -> (m = i+8, n)
  const int hi = lane >> 4;
#pragma unroll
  for (int i = 0; i < 8; ++i) {
    const int m = i + 8 * hi;
    const float y2 = __shfl(y2own, m, 32);   // lane m holds row m's norm
    const float s  = accS[i];                // <p_k, x_b>
    const float td = accT[i];                // <tan_k, x_b>

    const float alpha = 1.f - 2.f * s + y2;
    const float den   = fmaxf(1.f - 2.f * s + p2c * y2, EPSV);
    const float inv   = 1.f / den;

    float dn2 = (alpha * alpha * p2c - 2.f * alpha * beta * s +
                 beta * beta * y2) * inv * inv;
    dn2 = fmaxf(dn2, EPSV);

    const float sc = (beta * td - alpha * ptc) * inv;
    float d2 = (1.f - dn2) * anc;
    const float sgn = (d2 >= 0.f) ? 1.f : -1.f;
    d2 = sgn * fmaxf(fabsf(d2), EPSV);

    const float z = (2.f * sc) / d2;
    out[(r0 + m) * NUM_OUT + col] = asinhf(z) * anc;
  }
}

// ---------------------------------------------------------------------------
// Host-side launcher.
// Workspace: P[128*256] | P2[128] | PT[128] | AN[128]  (~132.6 KB fp32)
// ---------------------------------------------------------------------------
extern "C" void kernel_launch(void* const* d_in, const int* in_sizes, int n_in,
                              void* d_out, int out_size, void* d_ws, size_t ws_size,
                              hipStream_t stream) {
  const float* x       = (const float*)d_in[0];  // [2048, 256]
  const float* point   = (const float*)d_in[1];  // [128, 256]
  const float* tangent = (const float*)d_in[2];  // [128, 256]
  float* out = (float*)d_out;                    // [2048, 128]

  float* P  = (float*)d_ws;
  float* P2 = P + NUM_OUT * FEAT_DIM;
  float* PT = P2 + NUM_OUT;
  float* AN = PT + NUM_OUT;

  poincare_prep<<<NUM_OUT, FEAT_DIM, 0, stream>>>(point, tangent, P, P2, PT, AN);

  dim3 grid(BATCH / 16, NUM_OUT / 64);
  const size_t smem_bytes = (size_t)(16 + 64 + 64) * LDA * sizeof(float);
  poincare_mlr<<<grid, 128, smem_bytes, stream>>>(x, P, tangent, P2, PT, AN, out);
}